// CausalConv1d_74715251081247
// MI455X (gfx1250) — compile-verified
//
#include <hip/hip_runtime.h>
#include <cstdint>
#include <cstddef>

// ---------------------------------------------------------------------------
// Causal depthwise conv1d (K=4) + SiLU + conv-state update for MI455X/gfx1250.
// Memory-bound: ~515 MB of traffic, ~22us floor at 23.3 TB/s. No matrix op
// exists here (depthwise => no contraction dim), so the CDNA5 leverage is the
// data-movement path: b128 vector accesses, global_prefetch_b8, async
// global->LDS staging of the seq-id tile (ASYNCcnt), and NT store hints so
// the dead 256MB output stream doesn't evict x's halo reuse from the 192MB L2.
// ---------------------------------------------------------------------------

#define BLOCK   256      // 8 wave32s
#define CH_VEC  4        // channels per thread (float4)
#define TT      64       // tokens per block strip (halo overhead 3/64)
#define KTAPS   4

#if defined(__AMDGCN__) && \
    __has_builtin(__builtin_amdgcn_global_load_async_to_lds_b32) && \
    __has_builtin(__builtin_amdgcn_s_wait_asynccnt)
#define ASYNC_LDS_OK 1
typedef __attribute__((address_space(1))) int  gl_int;
typedef __attribute__((address_space(3))) int  lds_int;
#endif

// native clang vector (required by __builtin_nontemporal_store)
typedef float __attribute__((ext_vector_type(4))) nfloat4;

__device__ __forceinline__ float4 ld4(const float* p) {
    return *reinterpret_cast<const float4*>(p);
}

__device__ __forceinline__ void st4_nt(float* p, float4 v) {
    nfloat4 nv;
    nv.x = v.x; nv.y = v.y; nv.z = v.z; nv.w = v.w;
    __builtin_nontemporal_store(nv, reinterpret_cast<nfloat4*>(p));
}

__global__ __launch_bounds__(BLOCK) void conv_silu_kernel(
    const float* __restrict__ x,      // (T, D)
    const float* __restrict__ w,      // (D, K)
    const float* __restrict__ bias,   // (D,)
    const int*   __restrict__ seq,    // (T,)
    float*       __restrict__ out,    // (T, D)
    int T, int D)
{
    __shared__ int s_seq[TT + 3];

    const int g  = blockIdx.x * BLOCK + threadIdx.x;   // float4 column index
    const int t0 = blockIdx.y * TT;

    // ---- stage seq_idx[t0-3 .. t0+TT-1] into LDS (async path on gfx1250) ----
    {
        const int ti = threadIdx.x;
        if (ti < TT + 3) {
            const int t = t0 - 3 + ti;
#if defined(ASYNC_LDS_OK)
            if (t >= 0 && t < T) {
                __builtin_amdgcn_global_load_async_to_lds_b32(
                    (gl_int*)(seq + t),
                    (lds_int*)(&s_seq[ti]),
                    /*offset=*/0, /*cpol=*/0);
            } else {
                s_seq[ti] = -1;
            }
#else
            s_seq[ti] = (t >= 0 && t < T) ? seq[t] : -1;
#endif
        }
#if defined(ASYNC_LDS_OK)
        __builtin_amdgcn_s_wait_asynccnt(0);
#endif
        __syncthreads();
    }

    // ---- per-thread taps/bias: wtN = taps (j=0..3) for channel 4g+N ----
    const int c0 = g * CH_VEC;
    const float4 wt0 = ld4(w + (size_t)(c0 + 0) * KTAPS);
    const float4 wt1 = ld4(w + (size_t)(c0 + 1) * KTAPS);
    const float4 wt2 = ld4(w + (size_t)(c0 + 2) * KTAPS);
    const float4 wt3 = ld4(w + (size_t)(c0 + 3) * KTAPS);
    const float4 bv  = ld4(bias + c0);

    // ---- sliding register window: x[t-1], x[t-2], x[t-3] ----
    const float* xcol = x + c0;
    float*       ocol = out + c0;
    float4 xm1 = {0.f, 0.f, 0.f, 0.f};
    float4 xm2 = {0.f, 0.f, 0.f, 0.f};
    float4 xm3 = {0.f, 0.f, 0.f, 0.f};
    if (t0 - 1 >= 0) xm1 = ld4(xcol + (size_t)(t0 - 1) * D);
    if (t0 - 2 >= 0) xm2 = ld4(xcol + (size_t)(t0 - 2) * D);
    if (t0 - 3 >= 0) xm3 = ld4(xcol + (size_t)(t0 - 3) * D);

    for (int i = 0; i < TT; ++i) {
        const int t = t0 + i;
        if (t >= T) break;

        const float4 xc = ld4(xcol + (size_t)t * D);

        // prefetch the row 16 tokens ahead (lowers to global_prefetch_b8)
        if (i + 16 < TT)
            __builtin_prefetch(xcol + (size_t)(t + 16) * D, 0, 1);

        const int   s  = s_seq[i + 3];
        const float m1 = (s_seq[i + 2] == s) ? 1.f : 0.f;
        const float m2 = (s_seq[i + 1] == s) ? 1.f : 0.f;
        const float m3 = (s_seq[i + 0] == s) ? 1.f : 0.f;

        // shift 0 -> tap j=3 (.w), shift 1 -> j=2 (.z), shift 2 -> j=1 (.y),
        // shift 3 -> j=0 (.x)
        float4 a;
        a.x = bv.x + wt0.w * xc.x + m1 * wt0.z * xm1.x + m2 * wt0.y * xm2.x + m3 * wt0.x * xm3.x;
        a.y = bv.y + wt1.w * xc.y + m1 * wt1.z * xm1.y + m2 * wt1.y * xm2.y + m3 * wt1.x * xm3.y;
        a.z = bv.z + wt2.w * xc.z + m1 * wt2.z * xm1.z + m2 * wt2.y * xm2.z + m3 * wt2.x * xm3.z;
        a.w = bv.w + wt3.w * xc.w + m1 * wt3.z * xm1.w + m2 * wt3.y * xm2.w + m3 * wt3.x * xm3.w;

        // SiLU
        a.x = a.x / (1.f + __expf(-a.x));
        a.y = a.y / (1.f + __expf(-a.y));
        a.z = a.z / (1.f + __expf(-a.z));
        a.w = a.w / (1.f + __expf(-a.w));

        // NT store: output is never re-read; keep L2 for x halo reuse
        st4_nt(ocol + (size_t)t * D, a);

        xm3 = xm2; xm2 = xm1; xm1 = xc;
    }
}

// new_conv_state[n,d,:] = (exists b: state_ids[b]==n) ? x[conv_idx[b,:], d]
//                                                     : conv_state[n,d,:]
__global__ __launch_bounds__(BLOCK) void conv_state_kernel(
    const float* __restrict__ x,          // (T, D)
    const float* __restrict__ conv_state, // (NCACHE, D, K)
    const int*   __restrict__ conv_idx,   // (B, K)
    const int*   __restrict__ state_ids,  // (B,)
    float*       __restrict__ out_state,  // (NCACHE, D, K)
    int B, int NCACHE, int D)
{
    const int idx = blockIdx.x * BLOCK + threadIdx.x;  // over NCACHE*D rows
    if (idx >= NCACHE * D) return;
    const int n = idx / D;
    const int d = idx - n * D;

    int bsel = -1;
    for (int b = 0; b < B; ++b)
        if (state_ids[b] == n) bsel = b;   // last match wins (JAX .at[].set)

    float4 v;
    if (bsel >= 0) {
        v.x = x[(size_t)conv_idx[bsel * KTAPS + 0] * D + d];
        v.y = x[(size_t)conv_idx[bsel * KTAPS + 1] * D + d];
        v.z = x[(size_t)conv_idx[bsel * KTAPS + 2] * D + d];
        v.w = x[(size_t)conv_idx[bsel * KTAPS + 3] * D + d];
    } else {
        v = ld4(conv_state + (size_t)idx * KTAPS);
    }
    st4_nt(out_state + (size_t)idx * KTAPS, v);
}

extern "C" void kernel_launch(void* const* d_in, const int* in_sizes, int n_in,
                              void* d_out, int out_size, void* d_ws, size_t ws_size,
                              hipStream_t stream) {
    const float* x          = (const float*)d_in[0];   // (1, T, D)
    const float* weight     = (const float*)d_in[1];   // (D, 1, K) contiguous = (D, K)
    const float* bias       = (const float*)d_in[2];   // (D,)
    const float* conv_state = (const float*)d_in[3];   // (NCACHE, D, K)
    const int*   seq_idx    = (const int*)d_in[4];     // (T,)
    const int*   conv_idx   = (const int*)d_in[5];     // (B, K)
    const int*   state_ids  = (const int*)d_in[6];     // (B,)

    const int T      = in_sizes[4];
    const int D      = in_sizes[2];
    const int B      = in_sizes[6];
    const int NCACHE = in_sizes[3] / (D * KTAPS);

    float* out       = (float*)d_out;
    float* out_state = out + (size_t)T * D;

    // Main conv+SiLU: grid.x over channel-vec columns, grid.y over token strips
    dim3 grid(D / (BLOCK * CH_VEC), (T + TT - 1) / TT, 1);
    conv_silu_kernel<<<grid, dim3(BLOCK, 1, 1), 0, stream>>>(
        x, weight, bias, seq_idx, out, T, D);

    // Conv-state update (copy-or-gather, writes every output row)
    const int rows = NCACHE * D;
    conv_state_kernel<<<(rows + BLOCK - 1) / BLOCK, dim3(BLOCK, 1, 1), 0, stream>>>(
        x, conv_state, conv_idx, state_ids, out_state, B, NCACHE, D);
}